// CZRing_11355893531275
// MI455X (gfx1250) — compile-verified
//
#include <hip/hip_runtime.h>

// CZ-ring diagonal sign flip on a [8192, 4096] complex batch.
// Pure streaming kernel: 512 MiB total traffic, HBM-roofline-bound
// (~22.5 us at 23.3 TB/s). No matrix ops exist in this workload, so the
// optimal CDNA5 path is wide, fully-coalesced VMEM with NT temporal hints,
// not WMMA. Sign flip is done as a sign-bit XOR (no FP math).

typedef unsigned int v2u __attribute__((ext_vector_type(2)));
typedef unsigned int v4u __attribute__((ext_vector_type(4)));

#define BATCH_LOG2 12   // BATCH = 4096
#define NW_TOP     12   // n_wires - 1: MSB bit position of the row index
#define UNROLL     4
#define TPB        256

// One "unit" = 2 consecutive elements of one row.
//   input : two b64 loads (x_real, x_imag), lanes contiguous -> 256B/wave/instr
//   output: one b128 store of {r0,i0,r1,i1},  lanes contiguous -> 512B/wave/instr
__global__ __launch_bounds__(TPB) void cz_ring_apply(
    const v2u* __restrict__ xr,
    const v2u* __restrict__ xi,
    v4u* __restrict__ out,
    unsigned nunits)
{
    const unsigned lane      = threadIdx.x;
    const unsigned blockBase = blockIdx.x * (TPB * UNROLL);

#pragma unroll
    for (int k = 0; k < UNROLL; ++k) {
        const unsigned u = blockBase + (unsigned)k * TPB + lane;
        if (u < nunits) {
            // element base = 2*u ; row = (2*u) >> 12 = u >> 11
            const unsigned row = u >> (BATCH_LOG2 - 1);

            // ring-CZ parity: adjacent-bit pairs (popcount of row & row>>1
            // covers all 12 neighbor pairs of a 13-bit index) + wrap pair
            // (MSB & LSB).  sign flip == XOR of the float sign bit.
            const unsigned adj  = row & (row >> 1);
            const unsigned par  = __popc(adj) + ((row >> NW_TOP) & row & 1u);
            const unsigned flip = (par & 1u) << 31;

            const v2u r = __builtin_nontemporal_load(xr + u);
            const v2u i = __builtin_nontemporal_load(xi + u);

            v4u o;
            o.x = r.x ^ flip;   // real 0
            o.y = i.x ^ flip;   // imag 0
            o.z = r.y ^ flip;   // real 1
            o.w = i.y ^ flip;   // imag 1
            __builtin_nontemporal_store(o, out + u);
        }
    }
}

extern "C" void kernel_launch(void* const* d_in, const int* in_sizes, int n_in,
                              void* d_out, int out_size, void* d_ws, size_t ws_size,
                              hipStream_t stream) {
    const float* xr = (const float*)d_in[0];   // x_real [DIM, BATCH] f32
    const float* xi = (const float*)d_in[1];   // x_imag [DIM, BATCH] f32

    const unsigned total  = (unsigned)in_sizes[0];   // DIM*BATCH = 2^25
    const unsigned nunits = total >> 1;              // 2 elements per unit
    const unsigned blocks = (nunits + TPB * UNROLL - 1) / (TPB * UNROLL);

    cz_ring_apply<<<blocks, TPB, 0, stream>>>(
        (const v2u*)xr, (const v2u*)xi, (v4u*)d_out, nunits);
}